// FreqFusion_81192061763894
// MI455X (gfx1250) — compile-verified
//
#include <hip/hip_runtime.h>

typedef __attribute__((ext_vector_type(2))) float v2f;
typedef __attribute__((ext_vector_type(8))) float v8f;

#define WPB 4  // waves per block for WMMA kernels (128 threads)

// ---------------------------------------------------------------------
// CDNA5 async global->LDS helpers (ASYNCcnt path).  Inline asm per the
// bridge doc so it works on both toolchains.
// ---------------------------------------------------------------------
__device__ __forceinline__ unsigned lds_offset_of(const void* p) {
  // generic -> LDS(addrspace 3) cast, then ptrtoint gives the 32-bit LDS
  // byte offset that GLOBAL_LOAD_ASYNC_TO_LDS expects in its VDST VGPR.
  return (unsigned)(size_t)(const __attribute__((address_space(3))) char*)p;
}

__device__ __forceinline__ void async_load_b32(unsigned lds_off, const float* g) {
  asm volatile("global_load_async_to_lds_b32 %0, %1, off"
               :: "v"(lds_off), "v"((unsigned long long)(size_t)g)
               : "memory");
}

__device__ __forceinline__ void wait_async0() {
#if __has_builtin(__builtin_amdgcn_s_wait_asynccnt)
  __builtin_amdgcn_s_wait_asynccnt(0);
#else
  asm volatile("s_wait_asynccnt 0" ::: "memory");
#endif
}

// =====================================================================
// 1x1 conv as GEMM:  Y[b,co,p] = sum_ci W[co,ci] * X[b,ci,p] + bias[co]
// One wave computes a 16(co) x 16(pixel) tile with V_WMMA_F32_16X16X4_F32.
// A (16x4 f32) layout (ISA 7.12.2): lanes0-15 M=0..15; v0 holds K={0|2},
// v1 holds K={1|3} selected by lane>>4.  B (4x16): N=lane&15, K rows split
// the same way.  C/D: VGPR r -> M = r + 8*(lane>>4), N = lane&15.
// =====================================================================
__global__ __launch_bounds__(WPB * 32)
void conv1x1_wmma(const float* __restrict__ X, const float* __restrict__ W,
                  const float* __restrict__ bias, float* __restrict__ Y,
                  int Cin, int Cout, int HW) {
  const int lane = threadIdx.x & 31;
  const int wv   = threadIdx.x >> 5;
  const int half = lane >> 4;
  const int l16  = lane & 15;
  const int n0   = (blockIdx.x * WPB + wv) * 16;
  const int co0  = blockIdx.y * 16;
  const int b    = blockIdx.z;

  const float* Wp = W + (size_t)(co0 + l16) * Cin + half * 2;
  const float* Xp = X + (size_t)b * Cin * HW + (size_t)(half * 2) * HW + n0 + l16;

  v8f acc = {};
  for (int k = 0; k < Cin; k += 4) {
    __builtin_prefetch(Xp + (size_t)8 * HW, 0, 1);
    v2f a;  a.x = Wp[0];   a.y = Wp[1];
    v2f bv; bv.x = Xp[0];  bv.y = Xp[HW];
    acc = __builtin_amdgcn_wmma_f32_16x16x4_f32(false, a, false, bv,
                                                (short)0, acc, false, false);
    Wp += 4;
    Xp += (size_t)4 * HW;
  }

  float* Yp = Y + (size_t)b * Cout * HW + (size_t)co0 * HW + n0 + l16;
#pragma unroll
  for (int r = 0; r < 8; ++r) {
    const int m = r + half * 8;
    Yp[(size_t)m * HW] = acc[r] + bias[co0 + m];
  }
}

// =====================================================================
// 3x3 conv (pad 1) as 9 shifted GEMM taps accumulated into one WMMA tile.
// Weights: (Cout, Cin, 3, 3).  Rows co >= Cout contribute zero (masked
// loads); stores masked to valid co.  Tiles of 16 pixels never cross a
// row because Wd % 16 == 0.
// =====================================================================
__global__ __launch_bounds__(WPB * 32)
void conv3x3_wmma(const float* __restrict__ X, const float* __restrict__ W,
                  const float* __restrict__ bias, float* __restrict__ Y,
                  int Cin, int Cout, int H, int Wd) {
  const int HW   = H * Wd;
  const int lane = threadIdx.x & 31;
  const int wv   = threadIdx.x >> 5;
  const int half = lane >> 4;
  const int l16  = lane & 15;
  const int n0   = (blockIdx.x * WPB + wv) * 16;
  const int co0  = blockIdx.y * 16;
  const int b    = blockIdx.z;

  const int y  = n0 / Wd;
  const int x  = (n0 % Wd) + l16;
  const int co = co0 + l16;
  const bool coOK = (co < Cout);

  const float* Xb = X + (size_t)b * Cin * HW;
  v8f acc = {};

#pragma unroll
  for (int dy = -1; dy <= 1; ++dy) {
#pragma unroll
    for (int dx = -1; dx <= 1; ++dx) {
      const int tap = (dy + 1) * 3 + (dx + 1);
      const int sy = y + dy;
      const int sx = x + dx;
      const bool ok = (sy >= 0) && (sy < H) && (sx >= 0) && (sx < Wd);
      const float* Xp = Xb + (size_t)(half * 2) * HW + (ok ? (sy * Wd + sx) : 0);
      const float* Wp = W + ((size_t)co * Cin + half * 2) * 9 + tap;
      for (int k = 0; k < Cin; k += 4) {
        v2f a;
        a.x = coOK ? Wp[0] : 0.0f;
        a.y = coOK ? Wp[9] : 0.0f;
        v2f bv;
        bv.x = ok ? Xp[0]  : 0.0f;
        bv.y = ok ? Xp[HW] : 0.0f;
        acc = __builtin_amdgcn_wmma_f32_16x16x4_f32(false, a, false, bv,
                                                    (short)0, acc, false, false);
        Wp += 4 * 9;
        Xp += (size_t)4 * HW;
      }
    }
  }

  float* Yp = Y + (size_t)b * Cout * HW + n0 + l16;
#pragma unroll
  for (int r = 0; r < 8; ++r) {
    const int m = r + half * 8;
    if (co0 + m < Cout)
      Yp[(size_t)(co0 + m) * HW] = acc[r] + bias[co0 + m];
  }
}

// =====================================================================
// kernel_normalizer: out_i = exp(v_i - max)*ham_i / sum_j exp(v_j - max)*ham_j
// (algebraically identical to softmax -> *ham -> renormalize)
// =====================================================================
template <int K>
__global__ __launch_bounds__(256)
void knorm_kernel(const float* __restrict__ M, float* __restrict__ O,
                  int HW, int total) {
  const float h5[5] = {0.08f, 0.54f, 1.0f, 0.54f, 0.08f};
  const float h3[3] = {0.08f, 1.0f, 0.08f};
  int idx = blockIdx.x * 256 + threadIdx.x;
  if (idx >= total) return;
  const int b = idx / HW, p = idx % HW;
  const int KK = K * K;
  const float* mp = M + (size_t)b * KK * HW + p;
  float* op = O + (size_t)b * KK * HW + p;

  float v[K * K];
  float mx = -3.4e38f;
#pragma unroll
  for (int i = 0; i < KK; ++i) {
    v[i] = mp[(size_t)i * HW];
    mx = fmaxf(mx, v[i]);
  }
  float s = 0.f;
#pragma unroll
  for (int i = 0; i < KK; ++i) {
    const float ham = (K == 5) ? (h5[i / 5] * h5[i % 5]) : (h3[i / 3] * h3[i % 3]);
    const float e = __expf(v[i] - mx) * ham;
    v[i] = e;
    s += e;
  }
  const float inv = 1.0f / s;
#pragma unroll
  for (int i = 0; i < KK; ++i) op[(size_t)i * HW] = v[i] * inv;
}

// =====================================================================
// CARAFE up=1, k=3, fused residual, 4 channels per thread (mask loads
// amortized 4x):  O = 2*X - sum_t m[t,y,x]*Xpad[y+dy,x+dx]
// =====================================================================
__global__ __launch_bounds__(256)
void carafe3_residual4(const float* __restrict__ X, const float* __restrict__ M,
                       float* __restrict__ O, int C, int H, int Wd, int total) {
  int idx = blockIdx.x * 256 + threadIdx.x;   // over (b, C/4, HW)
  if (idx >= total) return;
  const int HW = H * Wd;
  const int p  = idx % HW;
  const int cq = (idx / HW) % (C >> 2);
  const int b  = idx / (HW * (C >> 2));
  const int c0 = cq * 4;
  const int y = p / Wd, x = p % Wd;

  const float* mp = M + (size_t)b * 9 * HW + p;
  float m[9];
#pragma unroll
  for (int t = 0; t < 9; ++t) m[t] = mp[(size_t)t * HW];

#pragma unroll
  for (int cs = 0; cs < 4; ++cs) {
    const float* xp = X + ((size_t)b * C + c0 + cs) * HW;
    float acc = 0.f;
#pragma unroll
    for (int i = 0; i < 3; ++i) {
      const int sy = y + i - 1;
      const bool oy = (sy >= 0) && (sy < H);
#pragma unroll
      for (int j = 0; j < 3; ++j) {
        const int sx = x + j - 1;
        const float xv = (oy && sx >= 0 && sx < Wd) ? xp[sy * Wd + sx] : 0.f;
        acc = fmaf(m[i * 3 + j], xv, acc);
      }
    }
    O[((size_t)b * C + c0 + cs) * HW + p] = 2.f * xp[p] - acc;
  }
}

// =====================================================================
// CARAFE up=2, k=5 (+ optional fused addend), LDS-staged masks:
//   O[b,c,Y,X] = Add[b,c,Y,X] + sum_t m[b,t,Y,X]*Xpad[b,c,Y/2+i-2,X/2+j-2]
// One block handles one output row Y (all channels).  The 25 x W2 mask
// slab for that row is copied global->LDS once with async loads
// (ASYNCcnt) and reused by every channel.  Each thread produces two
// adjacent outputs (xo, xo+1) sharing one 5x5 source window.
// X: (b,C,h,w); M/O/Add at (.., 2h, 2w).  Requires 2*w <= 128.
// =====================================================================
__global__ __launch_bounds__(256)
void carafe5_up2_lds(const float* __restrict__ X, const float* __restrict__ M,
                     const float* __restrict__ Add, float* __restrict__ O,
                     int C, int h, int w) {
  const int W2  = 2 * w;
  const int HWo = 4 * h * w;                  // output pixels per channel
  const int Y = blockIdx.x;                   // output row, [0, 2h)
  const int b = blockIdx.y;
  const int yl = Y >> 1;
  const int tid = threadIdx.x;

  __shared__ float sm[25 * 128];              // mask slab for this row

  // ---- async stage: 25 * W2 dwords of mask for row Y ----
  const float* Mrow = M + (size_t)b * 25 * HWo + (size_t)Y * W2;
  const int nElem = 25 * W2;
  for (int e = tid; e < nElem; e += 256) {
    const int t = e / W2;
    const int xx = e - t * W2;
    async_load_b32(lds_offset_of(&sm[e]), Mrow + (size_t)t * HWo + xx);
  }
  wait_async0();
  __syncthreads();

  // ---- compute: 4 channels x 64 output-pairs per 256-thread block ----
  const int csub = tid >> 6;                  // 0..3
  const int xp   = tid & 63;                  // source column xl, [0, w)
  const int xo   = xp * 2;                    // output column base

  for (int cc = 0; cc < C; cc += 4) {
    const int c = cc + csub;
    if (c >= C) continue;
    const float* xs = X + ((size_t)b * C + c) * h * w;

    float win[25];
#pragma unroll
    for (int i = 0; i < 5; ++i) {
      const int sy = yl + i - 2;
      const bool oy = (sy >= 0) && (sy < h);
#pragma unroll
      for (int j = 0; j < 5; ++j) {
        const int sx = xp + j - 2;
        win[i * 5 + j] = (oy && sx >= 0 && sx < w) ? xs[sy * w + sx] : 0.f;
      }
    }

    const size_t obase = ((size_t)b * C + c) * HWo + (size_t)Y * W2;
    float a0 = 0.f, a1 = 0.f;
    if (Add) { a0 = Add[obase + xo]; a1 = Add[obase + xo + 1]; }
#pragma unroll
    for (int t = 0; t < 25; ++t) {
      const float wv = win[t];
      a0 = fmaf(sm[t * W2 + xo],     wv, a0);
      a1 = fmaf(sm[t * W2 + xo + 1], wv, a1);
    }
    O[obase + xo]     = a0;
    O[obase + xo + 1] = a1;
  }
}

// =====================================================================
// Host-side orchestration
// =====================================================================
extern "C" void kernel_launch(void* const* d_in, const int* in_sizes, int n_in,
                              void* d_out, int out_size, void* d_ws, size_t ws_size,
                              hipStream_t stream) {
  (void)in_sizes; (void)n_in; (void)out_size; (void)ws_size;
  const float* hr_feat = (const float*)d_in[0];  // (2,256,128,128)
  const float* lr_feat = (const float*)d_in[1];  // (2,256,64,64)
  const float* W_hr    = (const float*)d_in[2];  // (64,256,1,1)
  const float* b_hr    = (const float*)d_in[3];
  const float* W_lr    = (const float*)d_in[4];
  const float* b_lr    = (const float*)d_in[5];
  const float* W_enc   = (const float*)d_in[6];  // (25,64,3,3)
  const float* b_enc   = (const float*)d_in[7];
  const float* W_enc2  = (const float*)d_in[8];  // (9,64,3,3)
  const float* b_enc2  = (const float*)d_in[9];

  const int B = 2, CHR = 256, CC = 64;
  const int H = 128, Wd = 128, HWhr = H * Wd;     // 16384
  const int h = 64,  w = 64,  HWlr = h * w;       // 4096

  float* ws = (float*)d_ws;
  size_t o = 0;
  float* chf        = ws + o; o += (size_t)B * CC * HWhr;   // (2,64,128,128)
  float* chf2       = ws + o; o += (size_t)B * CC * HWhr;
  float* clf        = ws + o; o += (size_t)B * CC * HWlr;   // (2,64,64,64)
  float* mask_hr_hr = ws + o; o += (size_t)B * 9  * HWhr;
  float* m_hi       = ws + o; o += (size_t)B * 9  * HWhr;
  float* mask_lr_hr = ws + o; o += (size_t)B * 25 * HWhr;
  float* m_lo       = ws + o; o += (size_t)B * 25 * HWhr;
  float* t_enc_lr   = ws + o; o += (size_t)B * 25 * HWlr;
  float* mask_lr    = ws + o; o += (size_t)B * 25 * HWhr;
  float* t_enc2_lr  = ws + o; o += (size_t)B * 9  * HWlr;
  float* mask_hr    = ws + o; o += (size_t)B * 9  * HWhr;
  float* mask_hr_n  = ws + o; o += (size_t)B * 9  * HWhr;

  float* out_mask = (float*)d_out;                      // (2,25,128,128)
  float* out_hr   = out_mask + (size_t)B * 25 * HWhr;   // (2,256,128,128)
  float* out_lr   = out_hr   + (size_t)B * CHR * HWhr;  // (2,256,128,128)

  // 1-2: 1x1 convs (fp32 WMMA GEMM)
  conv1x1_wmma<<<dim3(HWhr / 16 / WPB, CC / 16, B), WPB * 32, 0, stream>>>(
      hr_feat, W_hr, b_hr, chf, CHR, CC, HWhr);
  conv1x1_wmma<<<dim3(HWlr / 16 / WPB, CC / 16, B), WPB * 32, 0, stream>>>(
      lr_feat, W_lr, b_lr, clf, CHR, CC, HWlr);

  // 3: mask_hr_hr = conv3x3(chf, W_enc2)
  conv3x3_wmma<<<dim3(HWhr / 16 / WPB, 1, B), WPB * 32, 0, stream>>>(
      chf, W_enc2, b_enc2, mask_hr_hr, CC, 9, H, Wd);

  // 4: m_hi = knorm3(mask_hr_hr)
  {
    int total = B * HWhr;
    knorm_kernel<3><<<(total + 255) / 256, 256, 0, stream>>>(mask_hr_hr, m_hi, HWhr, total);
  }

  // 5: chf2 = 2*chf - carafe3(chf, m_hi)
  {
    int total = B * (CC / 4) * HWhr;
    carafe3_residual4<<<(total + 255) / 256, 256, 0, stream>>>(chf, m_hi, chf2, CC, H, Wd, total);
  }

  // 6: mask_lr_hr = conv3x3(chf2, W_enc)
  conv3x3_wmma<<<dim3(HWhr / 16 / WPB, 2, B), WPB * 32, 0, stream>>>(
      chf2, W_enc, b_enc, mask_lr_hr, CC, 25, H, Wd);

  // 7: m_lo = knorm5(mask_lr_hr)
  {
    int total = B * HWhr;
    knorm_kernel<5><<<(total + 255) / 256, 256, 0, stream>>>(mask_lr_hr, m_lo, HWhr, total);
  }

  // 8: t_enc_lr = conv3x3(clf, W_enc)
  conv3x3_wmma<<<dim3(HWlr / 16 / WPB, 2, B), WPB * 32, 0, stream>>>(
      clf, W_enc, b_enc, t_enc_lr, CC, 25, h, w);

  // 9: mask_lr = mask_lr_hr + carafe5_up2(t_enc_lr, m_lo)
  carafe5_up2_lds<<<dim3(2 * h, B), 256, 0, stream>>>(
      t_enc_lr, m_lo, mask_lr_hr, mask_lr, 25, h, w);

  // 10: mask_lr_n (== m_lo2) = knorm5(mask_lr) -> first output
  {
    int total = B * HWhr;
    knorm_kernel<5><<<(total + 255) / 256, 256, 0, stream>>>(mask_lr, out_mask, HWhr, total);
  }

  // 11: t_enc2_lr = conv3x3(clf, W_enc2)
  conv3x3_wmma<<<dim3(HWlr / 16 / WPB, 1, B), WPB * 32, 0, stream>>>(
      clf, W_enc2, b_enc2, t_enc2_lr, CC, 9, h, w);

  // 12: mask_hr = mask_hr_hr + carafe5_up2(t_enc2_lr, m_lo2)
  carafe5_up2_lds<<<dim3(2 * h, B), 256, 0, stream>>>(
      t_enc2_lr, out_mask, mask_hr_hr, mask_hr, 9, h, w);

  // 13: lr_out = carafe5_up2(lr_feat, mask_lr_n) -> third output
  carafe5_up2_lds<<<dim3(2 * h, B), 256, 0, stream>>>(
      lr_feat, out_mask, nullptr, out_lr, CHR, h, w);

  // 14: mask_hr_n = knorm3(mask_hr)
  {
    int total = B * HWhr;
    knorm_kernel<3><<<(total + 255) / 256, 256, 0, stream>>>(mask_hr, mask_hr_n, HWhr, total);
  }

  // 15: hr_out = 2*hr_feat - carafe3(hr_feat, mask_hr_n) -> second output
  {
    int total = B * (CHR / 4) * HWhr;
    carafe3_residual4<<<(total + 255) / 256, 256, 0, stream>>>(
        hr_feat, mask_hr_n, out_hr, CHR, H, Wd, total);
  }
}